// GMLayer_15479062135371
// MI455X (gfx1250) — compile-verified
//
#include <hip/hip_runtime.h>
#include <math.h>

typedef __attribute__((ext_vector_type(2))) float v2f;
typedef __attribute__((ext_vector_type(4))) float v4f;
typedef __attribute__((ext_vector_type(8))) float v8f;

#define F_DIM 64
#define G_DIM 64
#define LDS_STRIDE 68   // 64 + 4 pad: row r -> bank shift 4r, conflict-free b64 A loads

// ---------------------------------------------------------------------------
// Stage 1: per-gaussian Cholesky L, Linv = L^-1, v = Linv*c, log-normalizer.
// One 64-thread block per gaussian. Cost is negligible vs. the main GEMM.
// ---------------------------------------------------------------------------
__global__ __launch_bounds__(64) void gm_cholinv_kernel(const float* __restrict__ cov,
                                                        const float* __restrict__ centers,
                                                        float* __restrict__ Linv,
                                                        float* __restrict__ vout,
                                                        float* __restrict__ cgout) {
  __shared__ float Lm[F_DIM][F_DIM + 1];
  __shared__ float Inv[F_DIM][F_DIM + 1];
  const int g = blockIdx.x;
  const int t = threadIdx.x;

  for (int i = 0; i < F_DIM; ++i) Lm[i][t] = cov[g * F_DIM * F_DIM + i * F_DIM + t];
  __syncthreads();

  // Right-looking Cholesky, thread t owns row t.
  for (int k = 0; k < F_DIM; ++k) {
    if (t == k) Lm[k][k] = sqrtf(Lm[k][k]);
    __syncthreads();
    if (t > k) Lm[t][k] /= Lm[k][k];
    __syncthreads();
    if (t > k) {
      const float ljk = Lm[t][k];
      for (int i = k + 1; i <= t; ++i) Lm[t][i] -= ljk * Lm[i][k];
    }
    __syncthreads();
  }

  if (t == 0) {
    float ld = 0.f;
    for (int i = 0; i < F_DIM; ++i) ld += logf(Lm[i][i]);
    // -0.5*(F*log(2pi) + 2*sum(log diag))
    cgout[g] = -0.5f * (64.0f * 1.8378770664093453f) - ld;
  }

  // Forward substitution: thread t computes column t of Linv (L * col = e_t).
  for (int i = 0; i < t; ++i) Inv[i][t] = 0.0f;
  Inv[t][t] = 1.0f / Lm[t][t];
  for (int i = t + 1; i < F_DIM; ++i) {
    float s = 0.f;
    for (int j = t; j < i; ++j) s += Lm[i][j] * Inv[j][t];
    Inv[i][t] = -s / Lm[i][i];
  }
  __syncthreads();

  for (int i = 0; i < F_DIM; ++i)
    Linv[(size_t)g * F_DIM * F_DIM + i * F_DIM + t] = Inv[i][t];

  // v = Linv * c  (thread t computes row t)
  float s = 0.f;
  for (int j = 0; j < F_DIM; ++j) s += Inv[t][j] * centers[g * F_DIM + j];
  vout[g * F_DIM + t] = s;
}

// ---------------------------------------------------------------------------
// Stage 2: u = Linv_g * x  via V_WMMA_F32_16X16X4_F32, maha = ||u - v_g||^2,
// out = exp(const_g - 0.5*maha). One wave per 16-point N-tile; B (x tile)
// register-resident across all 64 gaussians; Linv_g staged in padded LDS.
// ---------------------------------------------------------------------------
__global__ __launch_bounds__(256) void gm_pdf_kernel(const float* __restrict__ x,
                                                     const float* __restrict__ Linv,
                                                     const float* __restrict__ vws,
                                                     const float* __restrict__ cgws,
                                                     float* __restrict__ out) {
  __shared__ float Ls[F_DIM * LDS_STRIDE + F_DIM];
  const int tid   = threadIdx.x;
  const int wave  = tid >> 5;
  const int lane  = tid & 31;
  const int lmod  = lane & 15;
  const int lhalf = lane >> 4;
  const int ntile = blockIdx.x * 8 + wave;
  const int p0    = ntile * 16;                 // first point of this wave's tile
  const int vOff  = F_DIM * LDS_STRIDE;

  // Preload B operands: B[k] covers K = 4k..4k+3.
  // B layout (4x16, K x N): lane l, VGPR r -> K = 2*(l/16)+r, N = l%16.
  v2f B[16];
  const float* xrow = x + (size_t)(p0 + lmod) * F_DIM + 2 * lhalf;
#pragma unroll
  for (int k = 0; k < 16; ++k)
    B[k] = *(const v2f*)(xrow + 4 * k);

  for (int g = 0; g < G_DIM; ++g) {
    // Cooperative staged load of Linv_g (4096 floats) into padded LDS.
#pragma unroll
    for (int tq = 0; tq < 4; ++tq) {
      const int e   = (tid + tq * 256) * 4;     // 4-aligned, never crosses a row
      const int row = e >> 6, col = e & 63;
      const v4f d = *(const v4f*)(Linv + (size_t)g * 4096 + e);
      *(v4f*)&Ls[row * LDS_STRIDE + col] = d;
    }
    if (tid < F_DIM) Ls[vOff + tid] = vws[g * F_DIM + tid];
    if (g + 1 < G_DIM)                          // pull next gaussian into L2/L0
      __builtin_prefetch(Linv + (size_t)(g + 1) * 4096 + tid * 16, 0, 1);
    __syncthreads();

    // A layout (16x4, M x K): lane l, VGPR r -> M = l%16, K = 2*(l/16)+r.
    v8f acc0 = {}, acc1 = {}, acc2 = {}, acc3 = {};
#pragma unroll
    for (int k = 0; k < 16; ++k) {
      const int coff = 4 * k + 2 * lhalf;
      const v2f a0 = *(const v2f*)&Ls[( 0 + lmod) * LDS_STRIDE + coff];
      const v2f a1 = *(const v2f*)&Ls[(16 + lmod) * LDS_STRIDE + coff];
      const v2f a2 = *(const v2f*)&Ls[(32 + lmod) * LDS_STRIDE + coff];
      const v2f a3 = *(const v2f*)&Ls[(48 + lmod) * LDS_STRIDE + coff];
      acc0 = __builtin_amdgcn_wmma_f32_16x16x4_f32(false, a0, false, B[k], (short)0, acc0, false, false);
      acc1 = __builtin_amdgcn_wmma_f32_16x16x4_f32(false, a1, false, B[k], (short)0, acc1, false, false);
      acc2 = __builtin_amdgcn_wmma_f32_16x16x4_f32(false, a2, false, B[k], (short)0, acc2, false, false);
      acc3 = __builtin_amdgcn_wmma_f32_16x16x4_f32(false, a3, false, B[k], (short)0, acc3, false, false);
    }

    // C/D layout: VGPR r, lane l -> M = r + 8*(l/16), N = l%16.
    float partial = 0.f;
#pragma unroll
    for (int r = 0; r < 8; ++r) {
      const int mb = 8 * lhalf + r;
      const float d0 = acc0[r] - Ls[vOff +  0 + mb];
      const float d1 = acc1[r] - Ls[vOff + 16 + mb];
      const float d2 = acc2[r] - Ls[vOff + 32 + mb];
      const float d3 = acc3[r] - Ls[vOff + 48 + mb];
      partial += d0 * d0 + d1 * d1 + d2 * d2 + d3 * d3;
    }
    const float maha = partial + __shfl_xor(partial, 16, 32);
    const float val  = expf(cgws[g] - 0.5f * maha);
    if (lane < 16)
      out[(size_t)(p0 + lane) * G_DIM + g] = val;
    __syncthreads();                            // LDS reused next gaussian
  }
}

// ---------------------------------------------------------------------------
extern "C" void kernel_launch(void* const* d_in, const int* in_sizes, int n_in,
                              void* d_out, int out_size, void* d_ws, size_t ws_size,
                              hipStream_t stream) {
  const float* x       = (const float*)d_in[0];  // (8, 2048, 64) f32
  const float* centers = (const float*)d_in[1];  // (64, 64) f32
  const float* cov     = (const float*)d_in[2];  // (64, 64, 64) f32
  float* out = (float*)d_out;                    // (8, 2048, 64) f32

  float* Linv = (float*)d_ws;                       // 64*64*64 floats (1 MB)
  float* vws  = Linv + G_DIM * F_DIM * F_DIM;       // 64*64 floats
  float* cgws = vws + G_DIM * F_DIM;                // 64 floats

  hipLaunchKernelGGL(gm_cholinv_kernel, dim3(G_DIM), dim3(F_DIM), 0, stream,
                     cov, centers, Linv, vws, cgws);

  const int npoints = in_sizes[0] / F_DIM;       // 16384
  const int ntiles  = npoints / 16;              // 1024 (16 points per wave)
  hipLaunchKernelGGL(gm_pdf_kernel, dim3(ntiles / 8), dim3(256), 0, stream,
                     x, Linv, vws, cgws, out);
}